// LJ_23613730193758
// MI455X (gfx1250) — compile-verified
//
#include <hip/hip_runtime.h>
#include <stdint.h>

#define N_ATOMS  100000
#define M_SEG    1000
#define E_EDGES  4000000
#define ZMAX_C   100
#define CUTOFF2  16.0f
#define SIG2FAC  0.7937005259840998f   // 2^(-1/3): sigma^2 = r0^2 * 2^(-1/3)

#define TPB          256
#define EDGE_BLOCKS  1024
#define TDM_COUNT    1500              // elements of sFF staged by the TDM; rest via async loads

// Use CDNA5 async global->LDS / TDM paths on the device pass only.
#if defined(__HIP_DEVICE_COMPILE__) && defined(__gfx1250__)
#define USE_ASYNC_LDS 1
#else
#define USE_ASYNC_LDS 0
#endif

typedef unsigned int u32x4 __attribute__((ext_vector_type(4)));
typedef int          i32x8 __attribute__((ext_vector_type(8)));
typedef int          i32x4 __attribute__((ext_vector_type(4)));

// ---------------- workspace layout (floats) ----------------
// [0     , 1000 ) energy accum
// [1000  , 4000 ) FF accum (SoA: x | y | z, 1000 each)
// [4000  , 7000 ) grad accum (SoA: x | y | z, 1000 each)
// [7168  , 7168+4N) packed atoms float4 (x,y,z shifted; w = e_neg)
// then N ints: aux = (film<<31)|(mol<<8)|Z
#define ATOM4_OFF  7168
#define AUX_OFF    (ATOM4_OFF + 4 * N_ATOMS)

__global__ void zero_acc(float* __restrict__ acc) {
    int t = blockIdx.x * blockDim.x + threadIdx.x;
    if (t < 7 * M_SEG) acc[t] = 0.0f;
}

__global__ void prep_atoms(const float* __restrict__ e_negs,
                           const float* __restrict__ R,
                           const float* __restrict__ shift,
                           const int*   __restrict__ Z,
                           const int*   __restrict__ idx_m,
                           const int*   __restrict__ film,
                           float4* __restrict__ atom4,
                           int*    __restrict__ aux) {
    int n = blockIdx.x * blockDim.x + threadIdx.x;
    if (n >= N_ATOMS) return;
    int m = idx_m[n];
    int f = film[n];
    float fm = (float)f;
    float4 a;
    a.x = R[3*n+0] + shift[3*m+0] * fm;
    a.y = R[3*n+1] + shift[3*m+1] * fm;
    a.z = R[3*n+2] + shift[3*m+2] * fm;
    a.w = e_negs[n];
    atom4[n] = a;
    aux[n] = (int)(((unsigned)f << 31) | ((unsigned)m << 8) | ((unsigned)Z[n] & 255u));
}

// Pass 1: per-edge energy + film forces, accumulated in LDS, flushed with global atomics.
__global__ void __launch_bounds__(TPB) edge_pass1(
        const int*    __restrict__ idx_i, const int* __restrict__ idx_j,
        const float*  __restrict__ off,   const float* __restrict__ tbl,
        const float4* __restrict__ atom4, const int* __restrict__ aux,
        float* __restrict__ acc) {
    __shared__ float s[4 * M_SEG];   // E | FFx | FFy | FFz
    for (int t = threadIdx.x; t < 4 * M_SEG; t += TPB) s[t] = 0.0f;
    __syncthreads();

    const int stride = gridDim.x * TPB;
    for (int e = blockIdx.x * TPB + threadIdx.x; e < E_EDGES; e += stride) {
        int pf = e + stride;
        if (pf < E_EDGES) {
            __builtin_prefetch(&idx_i[pf], 0, 0);
            __builtin_prefetch(&idx_j[pf], 0, 0);
            __builtin_prefetch(&off[3 * pf], 0, 0);
        }
        int i = idx_i[e], j = idx_j[e];
        float4 ai = atom4[i];
        float4 aj = atom4[j];
        float rx = aj.x - ai.x + off[3*e+0];
        float ry = aj.y - ai.y + off[3*e+1];
        float rz = aj.z - ai.z + off[3*e+2];
        float d2 = rx*rx + ry*ry + rz*rz;
        if (d2 >= CUTOFF2) continue;   // pot and its grads are where()'d to 0

        unsigned xi = (unsigned)aux[i], xj = (unsigned)aux[j];
        int fi = (int)(xi >> 31), fj = (int)(xj >> 31);
        int mi = (int)((xi >> 8) & 1023u), mj = (int)((xj >> 8) & 1023u);
        int zi = (int)(xi & 255u), zj = (int)(xj & 255u);

        float r0  = tbl[((fi * 2 + fj) * ZMAX_C + zi) * ZMAX_C + zj];
        float s2  = r0 * r0 * SIG2FAC;
        float eps = 1.0f + fabsf(ai.w - aj.w);
        float inv = __builtin_amdgcn_rcpf(d2);
        float t6  = s2 * inv;
        float sr6 = t6 * t6 * t6;
        float sr12 = sr6 * sr6;

        unsafeAtomicAdd(&s[mi], 4.0f * eps * (sr12 - sr6));

        if (fi | fj) {
            float c1 = 24.0f * eps * (sr6 - 2.0f * sr12) * inv;
            float gx = c1 * rx, gy = c1 * ry, gz = c1 * rz;
            if (fi) {
                unsafeAtomicAdd(&s[    M_SEG + mi],  gx);
                unsafeAtomicAdd(&s[2 * M_SEG + mi],  gy);
                unsafeAtomicAdd(&s[3 * M_SEG + mi],  gz);
            }
            if (fj) {
                unsafeAtomicAdd(&s[    M_SEG + mj], -gx);
                unsafeAtomicAdd(&s[2 * M_SEG + mj], -gy);
                unsafeAtomicAdd(&s[3 * M_SEG + mj], -gz);
            }
        }
    }
    __syncthreads();
    for (int t = threadIdx.x; t < 4 * M_SEG; t += TPB) {
        float v = s[t];
        if (v != 0.0f) unsafeAtomicAdd(&acc[t], v);
    }
}

// Pass 2: Hessian-vector products; FF staged into LDS via the TDM (tensor_load_to_lds)
// for the first TDM_COUNT elements and CDNA5 async global->LDS loads for the rest.
__global__ void __launch_bounds__(TPB) edge_pass2(
        const int*    __restrict__ idx_i, const int* __restrict__ idx_j,
        const float*  __restrict__ off,   const float* __restrict__ tbl,
        const float4* __restrict__ atom4, const int* __restrict__ aux,
        float* __restrict__ acc) {
    // Single shared block => static LDS starts at offset 0: sFF at byte 0, sG at byte 12000.
    __shared__ float smem[6 * M_SEG];
    float* sFF = smem;                 // staged FF (x|y|z, 1000 each)
    float* sG  = smem + 3 * M_SEG;     // grad accum (x|y|z)
    const float* __restrict__ ffg = acc + M_SEG;

    for (int t = threadIdx.x; t < 3 * M_SEG; t += TPB) {
        sG[t] = 0.0f;
#if USE_ASYNC_LDS
        if (t >= TDM_COUNT) {
            unsigned lds_off = 4u * (unsigned)t;   // byte offset of sFF[t] in LDS
            unsigned long long ga = (unsigned long long)(uintptr_t)(ffg + t);
            asm volatile("global_load_async_to_lds_b32 %0, %1, off"
                         :: "v"(lds_off), "v"(ga) : "memory");
        }
#else
        sFF[t] = ffg[t];
#endif
    }
#if USE_ASYNC_LDS
    if (threadIdx.x < 32) {   // one wave issues the TDM descriptor for sFF[0 .. TDM_COUNT)
        unsigned long long ga = (unsigned long long)(uintptr_t)ffg;
        u32x4 g0;
        g0.x = 1u;                                            // count=1 (valid), user mode
        g0.y = 0u;                                            // lds_addr = 0 (sFF base)
        g0.z = (unsigned)(ga & 0xffffffffull);                // global_addr[31:0]
        g0.w = (unsigned)((ga >> 32) & 0x01ffffffull)         // global_addr[56:32]
             | (2u << 30);                                    // type = 2 ("image")
        i32x8 g1;
        g1[0] = (int)(2u << 16);                              // data_size=2 -> 4 bytes
        g1[1] = (int)(((unsigned)TDM_COUNT & 0xffffu) << 16); // tensor_dim0[15:0] @ bits 63:48
        g1[2] = (int)(1u << 16);                              // tensor_dim1=1 @ bits 95:80
        g1[3] = (int)(((unsigned)TDM_COUNT & 0xffffu) << 16); // tile_dim0 @ bits 127:112
        g1[4] = 1;                                            // tile_dim1=1, tile_dim2=0
        g1[5] = TDM_COUNT;                                    // tensor_dim0_stride[31:0]
        g1[6] = 0;
        g1[7] = 0;
        i32x4 gz = {0, 0, 0, 0};
#if defined(__clang_major__) && (__clang_major__ >= 23)
        i32x8 gz8 = {0, 0, 0, 0, 0, 0, 0, 0};
        __builtin_amdgcn_tensor_load_to_lds(g0, g1, gz, gz, gz8, 0);
#else
        __builtin_amdgcn_tensor_load_to_lds(g0, g1, gz, gz, 0);
#endif
        __builtin_amdgcn_s_wait_tensorcnt(0);
    }
    asm volatile("s_wait_asynccnt 0x0" ::: "memory");
#endif
    __syncthreads();

    const int stride = gridDim.x * TPB;
    for (int e = blockIdx.x * TPB + threadIdx.x; e < E_EDGES; e += stride) {
        int pf = e + stride;
        if (pf < E_EDGES) {
            __builtin_prefetch(&idx_i[pf], 0, 0);
            __builtin_prefetch(&idx_j[pf], 0, 0);
            __builtin_prefetch(&off[3 * pf], 0, 0);
        }
        int i = idx_i[e], j = idx_j[e];
        unsigned xi = (unsigned)aux[i], xj = (unsigned)aux[j];
        int fi = (int)(xi >> 31), fj = (int)(xj >> 31);
        if ((fi | fj) == 0) continue;   // edge touches no film atom -> no contribution

        float4 ai = atom4[i];
        float4 aj = atom4[j];
        float rx = aj.x - ai.x + off[3*e+0];
        float ry = aj.y - ai.y + off[3*e+1];
        float rz = aj.z - ai.z + off[3*e+2];
        float d2 = rx*rx + ry*ry + rz*rz;
        if (d2 >= CUTOFF2) continue;

        int mi = (int)((xi >> 8) & 1023u), mj = (int)((xj >> 8) & 1023u);
        int zi = (int)(xi & 255u), zj = (int)(xj & 255u);

        float r0  = tbl[((fi * 2 + fj) * ZMAX_C + zi) * ZMAX_C + zj];
        float s2  = r0 * r0 * SIG2FAC;
        float eps = 1.0f + fabsf(ai.w - aj.w);
        float inv = __builtin_amdgcn_rcpf(d2);
        float t6  = s2 * inv;
        float sr6 = t6 * t6 * t6;
        float sr12 = sr6 * sr6;

        float c1 = 24.0f * eps * (sr6 - 2.0f * sr12) * inv;
        float c2 = 24.0f * eps * (28.0f * sr12 - 8.0f * sr6) * inv * inv;

        float fim = fi ? 2.0f : 0.0f;
        float fjm = fj ? 2.0f : 0.0f;
        float wx = fim * sFF[            mi] - fjm * sFF[            mj];
        float wy = fim * sFF[    M_SEG + mi] - fjm * sFF[    M_SEG + mj];
        float wz = fim * sFF[2 * M_SEG + mi] - fjm * sFF[2 * M_SEG + mj];

        float rw = rx * wx + ry * wy + rz * wz;
        float hx = c1 * wx + c2 * rw * rx;
        float hy = c1 * wy + c2 * rw * ry;
        float hz = c1 * wz + c2 * rw * rz;

        if (fj) {
            unsafeAtomicAdd(&sG[            mj],  hx);
            unsafeAtomicAdd(&sG[    M_SEG + mj],  hy);
            unsafeAtomicAdd(&sG[2 * M_SEG + mj],  hz);
        }
        if (fi) {
            unsafeAtomicAdd(&sG[            mi], -hx);
            unsafeAtomicAdd(&sG[    M_SEG + mi], -hy);
            unsafeAtomicAdd(&sG[2 * M_SEG + mi], -hz);
        }
    }
    __syncthreads();
    float* gacc = acc + 4 * M_SEG;
    for (int t = threadIdx.x; t < 3 * M_SEG; t += TPB) {
        float v = sG[t];
        if (v != 0.0f) unsafeAtomicAdd(&gacc[t], v);
    }
}

// out layout: [0,M) energy ; [M,2M) |FF|^2 ; [2M,5M) grad row-major (M,3)
__global__ void finalize(const float* __restrict__ acc, float* __restrict__ out) {
    int m = blockIdx.x * blockDim.x + threadIdx.x;
    if (m >= M_SEG) return;
    out[m] = acc[m];
    float fx = acc[    M_SEG + m];
    float fy = acc[2 * M_SEG + m];
    float fz = acc[3 * M_SEG + m];
    out[M_SEG + m] = fx * fx + fy * fy + fz * fz;
    out[2 * M_SEG + 3 * m + 0] = acc[4 * M_SEG + m];
    out[2 * M_SEG + 3 * m + 1] = acc[5 * M_SEG + m];
    out[2 * M_SEG + 3 * m + 2] = acc[6 * M_SEG + m];
}

extern "C" void kernel_launch(void* const* d_in, const int* in_sizes, int n_in,
                              void* d_out, int out_size, void* d_ws, size_t ws_size,
                              hipStream_t stream) {
    const float* e_negs  = (const float*)d_in[0];
    const float* R       = (const float*)d_in[1];
    const float* shift   = (const float*)d_in[2];
    const float* offsets = (const float*)d_in[3];
    const float* r0tab   = (const float*)d_in[4];
    const int*   Z       = (const int*)d_in[5];
    const int*   idx_m   = (const int*)d_in[6];
    const int*   idx_i   = (const int*)d_in[7];
    const int*   idx_j   = (const int*)d_in[8];
    const int*   is_film = (const int*)d_in[9];

    float*  ws    = (float*)d_ws;           // needs ~2.03 MB of scratch
    float*  acc   = ws;
    float4* atom4 = (float4*)(ws + ATOM4_OFF);
    int*    aux   = (int*)(ws + AUX_OFF);
    float*  out   = (float*)d_out;

    zero_acc<<<(7 * M_SEG + TPB - 1) / TPB, TPB, 0, stream>>>(acc);
    prep_atoms<<<(N_ATOMS + TPB - 1) / TPB, TPB, 0, stream>>>(
        e_negs, R, shift, Z, idx_m, is_film, atom4, aux);
    edge_pass1<<<EDGE_BLOCKS, TPB, 0, stream>>>(
        idx_i, idx_j, offsets, r0tab, atom4, aux, acc);
    edge_pass2<<<EDGE_BLOCKS, TPB, 0, stream>>>(
        idx_i, idx_j, offsets, r0tab, atom4, aux, acc);
    finalize<<<(M_SEG + TPB - 1) / TPB, TPB, 0, stream>>>(acc, out);
}